// Model_55233279426948
// MI455X (gfx1250) — compile-verified
//
#include <hip/hip_runtime.h>
#include <hip/hip_bf16.h>
#include <math.h>

typedef __attribute__((ext_vector_type(2))) float v2f;
typedef __attribute__((ext_vector_type(8))) float v8f;

#define NB 1024              // batch
#define MN 1024              // 32*32 pixels
#define PATCH_ROWS (49 * 1024)   // 50176
#define FEAT_ROWS  (81 * 256)    // 20736

// ---------------------------------------------------------------------------
// Kernel 1: per-image 7x7 zero-padded box sums as two banded-matrix GEMMs on
// the matrix core:  R = Ipad_h(32x40) * H(40x32),  S = V(32x40) * Rpad(40x32)
// where H[k][j] = (j<=k<=j+6), V[i][k] = (i<=k<=i+6). K padded 38->40 (pad
// columns multiply against all-zero band entries). One wave per image.
// LDS holds Ipad_h / Rpad with zero guard regions so fragment loads are
// unconditional ds_loads (no exec-mask branching around the WMMA pipeline).
// A-frag layout (f32 16x4): lanes 0-15: M=lane, K=0/1 in v0/v1;
//                           lanes16-31: M=lane-16, K=2/3 in v0/v1.
// B-frag layout (4x16):     v0: K = (lane<16?0:2), v1: K = (lane<16?1:3), N=lane&15.
// C/D layout:               VGPR v: lanes0-15 M=v, lanes16-31 M=v+8, N=lane&15.
// ---------------------------------------------------------------------------
__global__ __launch_bounds__(32) void k_boxsum_wmma(const float* __restrict__ img,
                                                    float* __restrict__ box) {
  __shared__ float simg[32][44];   // cols 0..39 = Ipad_h[i][k]; img in cols 3..34
  __shared__ float srow[40][36];   // rows 0..39 = Rpad[k][j];  R  in rows 3..34
  const int b = blockIdx.x;
  const int lane = threadIdx.x;

  // zero-fill (guard regions must be 0; zero everything, then overwrite)
  float* s0 = &simg[0][0];
  for (int idx = lane; idx < 32 * 44; idx += 32) s0[idx] = 0.0f;
  float* s1 = &srow[0][0];
  for (int idx = lane; idx < 40 * 36; idx += 32) s1[idx] = 0.0f;

  const float* ip = img + (size_t)b * MN;
#pragma unroll
  for (int r = 0; r < 32; ++r) simg[r][3 + lane] = ip[r * 32 + lane];
  __syncthreads();

  const int m = lane & 15;
  const int kh = (lane >> 4) << 1;    // K offset within k-step: 0 or 2
  const int rhalf = (lane >> 4) << 3; // C/D row half: 0 or 8

  // Pass A: horizontal 7-tap sums, R written into srow rows 3..34
#pragma unroll
  for (int Mt = 0; Mt < 2; ++Mt) {
#pragma unroll
    for (int Nt = 0; Nt < 2; ++Nt) {
      union { v8f v; float f[8]; } acc = {};
      const int row = Mt * 16 + m;
      const int n = Nt * 16 + m;
#pragma unroll
      for (int ks = 0; ks < 10; ++ks) {
        const int k0 = ks * 4 + kh;
        v2f a, h;
        a.x = simg[row][k0];
        a.y = simg[row][k0 + 1];
        h.x = ((unsigned)(k0 - n) <= 6u) ? 1.0f : 0.0f;       // n<=k0<=n+6
        h.y = ((unsigned)(k0 + 1 - n) <= 6u) ? 1.0f : 0.0f;
        acc.v = __builtin_amdgcn_wmma_f32_16x16x4_f32(false, a, false, h,
                                                      (short)0, acc.v, false, false);
      }
      const int rb = Mt * 16 + rhalf;
#pragma unroll
      for (int vv = 0; vv < 8; ++vv) srow[3 + rb + vv][Nt * 16 + m] = acc.f[vv];
    }
  }
  __syncthreads();

  // Pass B: vertical 7-tap sums of R
  float* bp = box + (size_t)b * MN;
#pragma unroll
  for (int Mt = 0; Mt < 2; ++Mt) {
#pragma unroll
    for (int Nt = 0; Nt < 2; ++Nt) {
      union { v8f v; float f[8]; } acc = {};
      const int i = Mt * 16 + m;
      const int n = Nt * 16 + m;
#pragma unroll
      for (int ks = 0; ks < 10; ++ks) {
        const int k0 = ks * 4 + kh;
        v2f a, rr;
        a.x = ((unsigned)(k0 - i) <= 6u) ? 1.0f : 0.0f;       // i<=k0<=i+6
        a.y = ((unsigned)(k0 + 1 - i) <= 6u) ? 1.0f : 0.0f;
        rr.x = srow[k0][n];
        rr.y = srow[k0 + 1][n];
        acc.v = __builtin_amdgcn_wmma_f32_16x16x4_f32(false, a, false, rr,
                                                      (short)0, acc.v, false, false);
      }
      const int rb = Mt * 16 + rhalf;
      const int col = Nt * 16 + m;
#pragma unroll
      for (int vv = 0; vv < 8; ++vv) bp[(rb + vv) * 32 + col] = acc.f[vv];
    }
  }
}

// ---------------------------------------------------------------------------
// Kernel 2: type-agnostic [1024][1024] 32-bit transpose (LDS tiled).
// ---------------------------------------------------------------------------
__global__ __launch_bounds__(256) void k_transpose1024(const unsigned* __restrict__ src,
                                                       unsigned* __restrict__ dst) {
  __shared__ unsigned tile[32][33];
  const int tx = threadIdx.x, ty = threadIdx.y;   // 32 x 8
  const int x0 = blockIdx.x * 32, y0 = blockIdx.y * 32;
#pragma unroll
  for (int r = ty; r < 32; r += 8)
    tile[r][tx] = src[(size_t)(y0 + r) * 1024 + x0 + tx];
  __syncthreads();
#pragma unroll
  for (int r = ty; r < 32; r += 8)
    dst[(size_t)(x0 + r) * 1024 + y0 + tx] = tile[tx][r];
}

// ---------------------------------------------------------------------------
// Kernel 3: patch rows (store-bandwidth bound, ~200 MB). One block per output
// row; 256 threads x float4 = 4KB row, batch is the fast axis -> fully
// coalesced b128 loads/stores. The bounds test is block-uniform (scalar).
// ---------------------------------------------------------------------------
__global__ __launch_bounds__(256) void k_patch(const float4* __restrict__ imgT4,
                                               const float4* __restrict__ bsT4,
                                               float4* __restrict__ out4) {
  const int row = blockIdx.x;        // 0..50175 (uniform)
  const int t = threadIdx.x;         // batch/4 chunk: 0..255
  const int pidx = row >> 10;
  const int pix = row & 1023;
  const int i = pix >> 5, j = pix & 31;
  const int pj = pidx / 7, pi = pidx - pj * 7;
  const int y = i + pi - 3, x = j + pj - 3;

  float4 v = make_float4(0.0f, 0.0f, 0.0f, 0.0f);
  if ((unsigned)y < 32u && (unsigned)x < 32u)
    v = imgT4[(size_t)(y * 32 + x) * 256 + t];
  const float4 bs = bsT4[(size_t)pix * 256 + t];
  const float inv49 = 1.0f / 49.0f;
  v.x -= bs.x * inv49;
  v.y -= bs.y * inv49;
  v.z -= bs.z * inv49;
  v.w -= bs.w * inv49;
  out4[(size_t)row * 256 + t] = v;
}

// ---------------------------------------------------------------------------
// Kernel 4: block entropy. Block = (32 batches x 8). LDS histogram u32[32][257]
// (stride 257 -> conflict-free), LDS atomics for the 49 adds, entropy LUT,
// batch-coalesced stores.
// ---------------------------------------------------------------------------
__global__ __launch_bounds__(256) void k_entropy(const unsigned* __restrict__ codesT,
                                                 float* __restrict__ out) {
  __shared__ unsigned cnt[32][257];
  __shared__ float lut[64];
  const int lane = threadIdx.x;            // batch within group of 32
  const int ty = threadIdx.y;              // 0..7
  const int tid = ty * 32 + lane;
  const int blk = blockIdx.y;              // 0..80  (blk = bi*9 + bj)
  const int batch = blockIdx.x * 32 + lane;
  const int bi = blk / 9, bj = blk - bi * 9;
  const int row0 = bi * 3, col0 = bj * 3;

  unsigned* cflat = &cnt[0][0];
  for (int idx = tid; idx < 32 * 257; idx += 256) cflat[idx] = 0u;
  if (tid < 64) {
    const float p = (float)tid * (1.0f / 49.0f);
    lut[tid] = (tid > 0 && tid <= 49) ? -p * log2f(p) : 0.0f;
  }
  __syncthreads();

  for (int c = ty; c < 49; c += 8) {
    const int r = c / 7, cc = c - r * 7;
    const unsigned code =
        codesT[(size_t)((row0 + r) * 32 + (col0 + cc)) * NB + batch] & 255u;
    atomicAdd(&cnt[lane][code], 1u);
  }
  __syncthreads();

  const size_t base = (size_t)(PATCH_ROWS + blk * 256) * NB + batch;
  for (int bin = ty; bin < 256; bin += 8)
    out[base + (size_t)bin * NB] = lut[cnt[lane][bin]];
}

// ---------------------------------------------------------------------------
extern "C" void kernel_launch(void* const* d_in, const int* in_sizes, int n_in,
                              void* d_out, int out_size, void* d_ws, size_t ws_size,
                              hipStream_t stream) {
  const float* img = (const float*)d_in[0];       // [1024][32][32] f32
  const int* codes = (const int*)d_in[1];         // [1024][32][32] i32
  float* out = (float*)d_out;                     // [70912][1024] f32

  // Workspace layout (16 MB total):
  float* wsf = (float*)d_ws;
  float* bsT     = wsf;                                // [MN][NB] box sums, pixel-major
  float* imgT    = wsf + (size_t)MN * NB;              // [MN][NB] image, pixel-major
  float* boxRaw  = wsf + 2 * (size_t)MN * NB;          // [NB][MN] box sums, batch-major
  unsigned* codesT = (unsigned*)(wsf + 3 * (size_t)MN * NB); // [MN][NB]

  // 1) box sums via WMMA (batch-major)
  k_boxsum_wmma<<<NB, 32, 0, stream>>>(img, boxRaw);
  // 2) transposes -> pixel-major so all later accesses are batch-coalesced
  k_transpose1024<<<dim3(32, 32), dim3(32, 8), 0, stream>>>((const unsigned*)img,
                                                            (unsigned*)imgT);
  k_transpose1024<<<dim3(32, 32), dim3(32, 8), 0, stream>>>((const unsigned*)boxRaw,
                                                            (unsigned*)bsT);
  k_transpose1024<<<dim3(32, 32), dim3(32, 8), 0, stream>>>((const unsigned*)codes,
                                                            codesT);
  // 3) patch rows (store-bandwidth bound; fully coalesced b128)
  k_patch<<<PATCH_ROWS, 256, 0, stream>>>((const float4*)imgT, (const float4*)bsT,
                                          (float4*)out);
  // 4) entropy rows
  k_entropy<<<dim3(32, 81), dim3(32, 8), 0, stream>>>(codesT, out);
}